// ChamferLoss_70927089926224
// MI455X (gfx1250) — compile-verified
//
#include <hip/hip_runtime.h>

typedef float v2f __attribute__((ext_vector_type(2)));
typedef float v8f __attribute__((ext_vector_type(8)));

#define WAVES_PER_BLOCK 16
#define BLOCK_THREADS   (WAVES_PER_BLOCK * 32)

// ds_swizzle_b32 group-of-32 XOR shuffle: offset15=0, xor_mask<<10, and_mask=0x1F
#define SWZ_XOR(x, imm) \
  __int_as_float(__builtin_amdgcn_ds_swizzle(__float_as_int(x), (imm)))
#define SWZ_X1  0x041F
#define SWZ_X2  0x081F
#define SWZ_X4  0x101F
#define SWZ_X8  0x201F
#define SWZ_X16 0x401F

__global__ void chamfer_zero_out(float* out, int n) {
  int i = blockIdx.x * blockDim.x + threadIdx.x;
  if (i < n) out[i] = 0.0f;
}

// One block handles 16 "A" points (rows) x all M "B" points (columns) for one batch.
// Accumulates  sum_{m in 16 rows} min_n ||a_m - b_n||  into out[b] via one atomicAdd.
// d^2 = ||a||^2 + ||b||^2 - 2 a.b :
//   - (-2 a.b) tile computed by V_WMMA_F32_16X16X4_F32 (A pre-scaled by -2, K=3 padded to 4)
//   - ||b||^2 added per-lane (output column fixed per lane in C/D layout)
//   - ||a||^2 added after the min (constant per row); sqrt(min(d2)) == min(sqrt(d2))
__global__ __launch_bounds__(BLOCK_THREADS) void chamfer_dir(
    const float* __restrict__ Apts, const float* __restrict__ Bpts,
    float* __restrict__ out, int N, int M) {
  const int b = blockIdx.y;
  const int m0 = blockIdx.x * 16;          // first of this block's 16 A-rows
  const int lane = threadIdx.x & 31;
  // Wave index as a true SGPR so the tile loop is a scalar loop
  // (EXEC provably all-1s at every WMMA; no per-lane loop counter VALU).
  const int wave = __builtin_amdgcn_readfirstlane(threadIdx.x) >> 5;
  const int halfLane = lane & 15;
  const bool hi = lane >= 16;

  const float* Abase = Apts + (size_t)(b * N) * 3;
  const float* Bbase = Bpts + (size_t)(b * M) * 3;

  // A operand, ISA 16x4 f32 layout:
  //   lanes 0-15 : (K0,K1) = (-2x, -2y) of row m0+lane
  //   lanes 16-31: (K2,K3) = (-2z,  0 ) of row m0+lane-16
  const float* ap = Abase + (size_t)(m0 + halfLane) * 3;
  float ax = ap[0], ay = ap[1], az = ap[2];
  v2f amat;
  amat.x = hi ? (-2.0f * az) : (-2.0f * ax);
  amat.y = hi ? 0.0f : (-2.0f * ay);

  float rmin[8];
#pragma unroll
  for (int s = 0; s < 8; ++s) rmin[s] = __builtin_inff();

  const int numTiles = M >> 4;             // 512; 32 tiles per wave, scalar trip count
  for (int tc = wave; tc < numTiles; tc += WAVES_PER_BLOCK) {
    const int n = tc * 16 + halfLane;
    const float* qp = Bbase + (size_t)n * 3;
    // prefetch next tile's column points for this wave (speculative; OOB dropped)
    __builtin_prefetch(qp + (size_t)WAVES_PER_BLOCK * 16 * 3, 0, 3);
    float qx = qp[0], qy = qp[1], qz = qp[2];
    float yy = qx * qx + qy * qy + qz * qz;      // ||b_n||^2 (per-lane scalar)
    // B operand, 4x16 layout mirrored with A:
    v2f bmat;
    bmat.x = hi ? qz : qx;
    bmat.y = hi ? 0.0f : qy;
    v8f c = {0.f, 0.f, 0.f, 0.f, 0.f, 0.f, 0.f, 0.f};
    v8f d = __builtin_amdgcn_wmma_f32_16x16x4_f32(false, amat, false, bmat,
                                                  (short)0, c, false, false);
#pragma unroll
    for (int s = 0; s < 8; ++s) {
      float v = d[s] + yy;                       // ||b_n||^2 - 2 a_m . b_n
      rmin[s] = fminf(rmin[s], v);
    }
  }

  // Min across the 16 lane-columns within each half-wave (C/D: rows r / r+8 per half).
  // XOR masks 1,2,4,8 never cross the half boundary -> ds_swizzle group-of-32.
#pragma unroll
  for (int s = 0; s < 8; ++s) {
    rmin[s] = fminf(rmin[s], SWZ_XOR(rmin[s], SWZ_X1));
    rmin[s] = fminf(rmin[s], SWZ_XOR(rmin[s], SWZ_X2));
    rmin[s] = fminf(rmin[s], SWZ_XOR(rmin[s], SWZ_X4));
    rmin[s] = fminf(rmin[s], SWZ_XOR(rmin[s], SWZ_X8));
  }

  __shared__ float lds[WAVES_PER_BLOCK][16];
  if (lane == 0) {
#pragma unroll
    for (int s = 0; s < 8; ++s) lds[wave][s] = rmin[s];        // rows 0..7
  } else if (lane == 16) {
#pragma unroll
    for (int s = 0; s < 8; ++s) lds[wave][8 + s] = rmin[s];    // rows 8..15
  }
  __syncthreads();

  if (threadIdx.x < 32) {                  // wave 0 finalizes
    float dist = 0.0f;
    if (threadIdx.x < 16) {
      float v = __builtin_inff();
#pragma unroll
      for (int w = 0; w < WAVES_PER_BLOCK; ++w) v = fminf(v, lds[w][threadIdx.x]);
      const float* pp = Abase + (size_t)(m0 + threadIdx.x) * 3;
      float xx = pp[0] * pp[0] + pp[1] * pp[1] + pp[2] * pp[2];   // ||a_m||^2
      dist = sqrtf(fmaxf(v + xx, 0.0f));
    }
    // Sum across all 32 lanes (lanes 16-31 contribute 0).
    dist += SWZ_XOR(dist, SWZ_X1);
    dist += SWZ_XOR(dist, SWZ_X2);
    dist += SWZ_XOR(dist, SWZ_X4);
    dist += SWZ_XOR(dist, SWZ_X8);
    dist += SWZ_XOR(dist, SWZ_X16);
    if (threadIdx.x == 0) atomicAdd(out + b, dist);
  }
}

extern "C" void kernel_launch(void* const* d_in, const int* in_sizes, int n_in,
                              void* d_out, int out_size, void* d_ws, size_t ws_size,
                              hipStream_t stream) {
  (void)in_sizes; (void)n_in; (void)d_ws; (void)ws_size; (void)out_size;
  const float* gts   = (const float*)d_in[0];
  const float* preds = (const float*)d_in[1];
  float* out = (float*)d_out;
  const int B = 4, N = 8192, M = 8192;

  chamfer_zero_out<<<1, 64, 0, stream>>>(out, B);
  dim3 grid(N / 16, B);
  // loss_2: for each gt point, min over preds
  chamfer_dir<<<grid, BLOCK_THREADS, 0, stream>>>(gts, preds, out, N, M);
  // loss_1: for each pred point, min over gts
  chamfer_dir<<<grid, BLOCK_THREADS, 0, stream>>>(preds, gts, out, M, N);
}